// Kmeans_5686536700012
// MI455X (gfx1250) — compile-verified
//
#include <hip/hip_runtime.h>

typedef __attribute__((ext_vector_type(2))) float v2f;
typedef __attribute__((ext_vector_type(8))) float v8f;

#define DDIM 256
#define CTILE 16
#define CB_STRIDE 260   // 16-row centroid tile, padded to avoid 64-bank conflicts

// ---------------------------------------------------------------------------
// Phase 0a: zero scratch (csq, counts, sums are contiguous)
__global__ __launch_bounds__(256) void kmeans_zero(float* __restrict__ p, int n) {
    int i = blockIdx.x * 256 + threadIdx.x;
    if (i < n) p[i] = 0.0f;
}

// Phase 0b: c_sq[k] = sum_d centroids[k][d]^2
__global__ __launch_bounds__(256) void kmeans_csq(const float* __restrict__ cent,
                                                  float* __restrict__ csq, int K) {
    int k = blockIdx.x * 256 + threadIdx.x;
    if (k >= K) return;
    const float* c = cent + (size_t)k * DDIM;
    float s = 0.0f;
    for (int d = 0; d < DDIM; ++d) s += c[d] * c[d];
    csq[k] = s;
}

// ---------------------------------------------------------------------------
// Phase 1: assignment via V_WMMA_F32_16X16X4_F32.
// 8 waves / WG, each wave handles 16 points. A-fragments (x tile) live in
// VGPRs for the entire centroid sweep; centroid tiles double-buffered in LDS.
__global__ __launch_bounds__(256) void kmeans_assign(
    const float* __restrict__ x, const float* __restrict__ cent,
    const float* __restrict__ csq, int* __restrict__ assigns, int K) {
    __shared__ float cbuf[2][CTILE * CB_STRIDE];

    const int tid   = threadIdx.x;
    const int lane  = tid & 31;
    const int wave  = tid >> 5;          // 0..7
    const int nl    = lane & 15;         // N (or M) index within tile
    const int khalf = (lane >> 4) & 1;   // second K-half in lanes 16..31
    const int m0    = blockIdx.x * 128 + wave * 16;

    // A fragments: lane holds x[m0+nl][4j + 2*khalf + {0,1}] for j = 0..63
    v2f a[64];
    {
        const float* xrow = x + (size_t)(m0 + nl) * DDIM + 2 * khalf;
        #pragma unroll
        for (int j = 0; j < 64; ++j) {
            a[j].x = xrow[4 * j + 0];
            a[j].y = xrow[4 * j + 1];
        }
    }

    float bestVal[8];
    int   bestIdx[8];
    #pragma unroll
    for (int r = 0; r < 8; ++r) { bestVal[r] = 3.4e38f; bestIdx[r] = 0; }

    const int nctiles = K / CTILE;

    // stage centroid tile 0
    for (int i = tid; i < CTILE * DDIM; i += 256) {
        int row = i >> 8, col = i & (DDIM - 1);
        cbuf[0][row * CB_STRIDE + col] = cent[(size_t)row * DDIM + col];
    }

    for (int ct = 0; ct < nctiles; ++ct) {
        const int cur = ct & 1;
        __syncthreads();                         // staging of `cur` complete
        if (ct + 1 < nctiles) {                  // prefetch next tile
            const float* csrc = cent + (size_t)(ct + 1) * CTILE * DDIM;
            for (int i = tid; i < CTILE * DDIM; i += 256) {
                int row = i >> 8, col = i & (DDIM - 1);
                cbuf[1 - cur][row * CB_STRIDE + col] = csrc[(size_t)row * DDIM + col];
            }
        }

        // 16x16 dot tile: acc[m][n] = sum_d x[m][d] * c[n][d]
        v8f acc = {};
        const float* bb = &cbuf[cur][nl * CB_STRIDE + 2 * khalf];
        #pragma unroll
        for (int j = 0; j < 64; ++j) {
            v2f b;
            b.x = bb[4 * j + 0];
            b.y = bb[4 * j + 1];
            acc = __builtin_amdgcn_wmma_f32_16x16x4_f32(
                false, a[j], false, b, (short)0, acc, false, false);
        }

        const int   nglob = ct * CTILE + nl;
        const float cs    = csq[nglob];
        #pragma unroll
        for (int r = 0; r < 8; ++r) {
            float val = cs - 2.0f * acc[r];
            if (val < bestVal[r]) { bestVal[r] = val; bestIdx[r] = nglob; }
        }
    }

    // Reduce (min, argmin) across the 16-lane N group. C layout: VGPR r holds
    // point m0+r in lanes 0-15 and point m0+r+8 in lanes 16-31.
    #pragma unroll
    for (int r = 0; r < 8; ++r) {
        float v   = bestVal[r];
        int   idx = bestIdx[r];
        #pragma unroll
        for (int mask = 8; mask >= 1; mask >>= 1) {
            float ov = __shfl_xor(v, mask, 32);
            int   oi = __shfl_xor(idx, mask, 32);
            if (ov < v || (ov == v && oi < idx)) { v = ov; idx = oi; }
        }
        if (nl == 0) assigns[m0 + r + khalf * 8] = idx;
    }
}

// ---------------------------------------------------------------------------
// Phase 2: scatter-accumulate sums/counts. One point per wave; 32-lane
// coalesced loads, fp32 global atomics into L2.
__global__ __launch_bounds__(256) void kmeans_accum(
    const float* __restrict__ x, const int* __restrict__ assigns,
    float* __restrict__ sums, float* __restrict__ counts, int N) {
    int wave = threadIdx.x >> 5;
    int lane = threadIdx.x & 31;
    int p = blockIdx.x * 8 + wave;
    if (p >= N) return;
    int a = assigns[p];
    const float* xp = x + (size_t)p * DDIM;
    float* sp = sums + (size_t)a * DDIM;
    #pragma unroll
    for (int i = 0; i < DDIM / 32; ++i)
        atomicAdd(&sp[lane + 32 * i], xp[lane + 32 * i]);
    if (lane == 0) atomicAdd(&counts[a], 1.0f);
}

// ---------------------------------------------------------------------------
// Phase 3: means with empty-cluster fallback.
__global__ __launch_bounds__(256) void kmeans_final(
    const float* __restrict__ sums, const float* __restrict__ counts,
    const float* __restrict__ cent, float* __restrict__ out, int total) {
    int i = blockIdx.x * 256 + threadIdx.x;
    if (i >= total) return;
    int k = i >> 8;  // D == 256
    float c = counts[k];
    out[i] = (c > 0.0f) ? sums[i] / c : cent[i];
}

// ---------------------------------------------------------------------------
extern "C" void kernel_launch(void* const* d_in, const int* in_sizes, int n_in,
                              void* d_out, int out_size, void* d_ws, size_t ws_size,
                              hipStream_t stream) {
    (void)n_in; (void)out_size; (void)ws_size;
    const float* x    = (const float*)d_in[0];
    const float* cent = (const float*)d_in[1];
    float* out = (float*)d_out;

    const int N = in_sizes[0] / DDIM;   // 131072
    const int K = in_sizes[1] / DDIM;   // 1024

    // Workspace layout: csq[K] | counts[K] | sums[K*D] | assigns[N]
    float* csq    = (float*)d_ws;
    float* counts = csq + K;
    float* sums   = counts + K;
    int*   assigns = (int*)(sums + (size_t)K * DDIM);

    const int nzero = 2 * K + K * DDIM;
    kmeans_zero<<<(nzero + 255) / 256, 256, 0, stream>>>(csq, nzero);
    kmeans_csq<<<(K + 255) / 256, 256, 0, stream>>>(cent, csq, K);
    kmeans_assign<<<N / 128, 256, 0, stream>>>(x, cent, csq, assigns, K);
    kmeans_accum<<<(N + 7) / 8, 256, 0, stream>>>(x, assigns, sums, counts, N);
    kmeans_final<<<(K * DDIM + 255) / 256, 256, 0, stream>>>(sums, counts, cent, out, K * DDIM);
}